// CGCNNConv_2156073582916
// MI455X (gfx1250) — compile-verified
//
#include <hip/hip_runtime.h>
#include <math.h>

// ---------------- problem constants ----------------
#define N_NODES 100000
#define N_EDGES 1600000
#define NF      64
#define EF      32
#define CH      128          // 2*NF, edge-BN channels
#define BN_EPS  1e-5f

typedef float v2f __attribute__((ext_vector_type(2)));
typedef float v8f __attribute__((ext_vector_type(8)));

// address-space helper types for async global->LDS copies
// (builtin signature per hipcc diagnostic: AS(1) int* / AS(3) int*)
typedef __attribute__((address_space(1))) int* glb_i32p;
typedef __attribute__((address_space(3))) int* lds_i32p;

__device__ __forceinline__ lds_i32p to_lds(void* p) {
  // generic LDS pointer -> AS(3): hardware uses low 32 bits as LDS offset
  return (lds_i32p)(unsigned int)(uintptr_t)p;
}
__device__ __forceinline__ glb_i32p to_glb(const void* p) {
  return (glb_i32p)(uintptr_t)p;
}

// ---------------- workspace layout (bytes) ----------------
// Hcat : N_NODES x 256 f32  (cols 0..127 = h_src+b, 128..255 = h_dst+b)
#define OFF_HCAT   ((size_t)0)
#define SZ_HCAT    ((size_t)N_NODES * 256 * 4)          // 102,400,000
#define OFF_ACC    (OFF_HCAT + SZ_HCAT)
#define SZ_ACC     ((size_t)N_NODES * NF * 4)           // 25,600,000
#define OFF_ESTAT  (OFF_ACC + SZ_ACC)                   // 256 f32: s1[128], s2[128]
#define SZ_ESTAT   ((size_t)256 * 4)
#define OFF_EBN    (OFF_ESTAT + SZ_ESTAT)               // 256 f32: scale[128], shift[128]
#define SZ_EBN     ((size_t)256 * 4)
#define OFF_NSTAT  (OFF_EBN + SZ_EBN)                   // 128 f32: s1[64], s2[64]
#define SZ_NSTAT   ((size_t)128 * 4)
#define OFF_NBN    (OFF_NSTAT + SZ_NSTAT)               // 128 f32: scale[64], shift[64]

__device__ __forceinline__ float softplus_f(float x) {
  return (x > 20.f) ? x : log1pf(__expf(x));
}
__device__ __forceinline__ float sigmoid_f(float x) {
  return 1.f / (1.f + __expf(-x));
}

// =====================================================================
// Stage W_edge (32x128, row-major) into LDS transposed+padded:
// lds_wt[col*34 + k] = W[k][col].  Uses gfx1250 async global->LDS copies
// (per-lane LDS destination lets the async engine do the transpose
// scatter); falls back to plain loads if the builtin is unavailable.
// =====================================================================
__device__ __forceinline__ void stage_wedge(const float* __restrict__ Wedge,
                                            float* lds_wt) {
#if __has_builtin(__builtin_amdgcn_global_load_async_to_lds_b32)
  for (int i = threadIdx.x; i < EF * CH; i += blockDim.x) {
    const int k = i >> 7, col = i & 127;          // Wedge row-major [k][col]
    __builtin_amdgcn_global_load_async_to_lds_b32(
        to_glb(Wedge + i), to_lds(&lds_wt[col * 34 + k]),
        /*offset=*/0, /*cpol=*/0);
  }
  asm volatile("s_wait_asynccnt 0x0" ::: "memory");
#else
  for (int i = threadIdx.x; i < EF * CH; i += blockDim.x) {
    const int k = i >> 7, col = i & 127;
    lds_wt[col * 34 + k] = Wedge[i];
  }
#endif
}

// =====================================================================
// Kernel 1: Hcat[node][0:128]   = node_feats @ W_src + b_src
//           Hcat[node][128:256] = node_feats @ W_dst + b_dst
// One wave per 16x16 output tile. K=64 in steps of 4 (f32 WMMA).
// =====================================================================
__global__ void k_node_gemm(const float* __restrict__ nf,
                            const float* __restrict__ Wsrc, const float* __restrict__ bsrc,
                            const float* __restrict__ Wdst, const float* __restrict__ bdst,
                            float* __restrict__ Hcat) {
  const int lane = threadIdx.x & 31;
  const int wave = threadIdx.x >> 5;
  const int wid  = blockIdx.x * 8 + wave;   // 0 .. 99999
  const int ntile = wid & 15;               // 16 tiles over 256 cols
  const int mtile = wid >> 4;               // 6250 tiles over 100000 rows
  const int lo = lane & 15;
  const int hi = lane >> 4;

  const int row = mtile * 16 + lo;          // A row (node)
  const int col = ntile * 16 + lo;          // B/C col
  const bool is_dst = (col >= CH);
  const float* __restrict__ W = is_dst ? Wdst : Wsrc;
  const int wcol = is_dst ? (col - CH) : col;

  v8f c = {};
#pragma unroll
  for (int k0 = 0; k0 < NF; k0 += 4) {
    const int k = k0 + 2 * hi;
    v2f a = *(const v2f*)(nf + (size_t)row * NF + k);
    v2f b;
    b.x = W[(size_t)k * CH + wcol];
    b.y = W[(size_t)(k + 1) * CH + wcol];
    c = __builtin_amdgcn_wmma_f32_16x16x4_f32(false, a, false, b, (short)0, c,
                                              false, false);
  }
  const float bias = is_dst ? bdst[wcol] : bsrc[wcol];
#pragma unroll
  for (int v = 0; v < 8; ++v) {
    const int r = mtile * 16 + v + 8 * hi;
    Hcat[(size_t)r * 256 + col] = c[v] + bias;
  }
}

// =====================================================================
// Edge GEMM tile helper: acc[n] = 16x16 tile n of (edge_feats @ W_edge)
// for 16 edges starting at ebase.  W_edge staged transposed+padded in LDS.
// =====================================================================
__device__ __forceinline__ void edge_gemm_tile(const float* __restrict__ ef,
                                               const float* lds_wt,
                                               int ebase, int lo, int hi,
                                               v8f acc[8]) {
  // stream hint for a future block's edge-feature tile (global_prefetch_b8)
  __builtin_prefetch(ef + ((size_t)ebase + 2048) * EF, 0, 1);
#pragma unroll
  for (int n = 0; n < 8; ++n) acc[n] = (v8f){};
#pragma unroll
  for (int k0 = 0; k0 < EF; k0 += 4) {
    const int k = k0 + 2 * hi;
    v2f a = *(const v2f*)(ef + (size_t)(ebase + lo) * EF + k);
#pragma unroll
    for (int n = 0; n < 8; ++n) {
      const int col = n * 16 + lo;
      v2f b = *(const v2f*)(lds_wt + col * 34 + k);   // pad 34 kills bank conflicts
      acc[n] = __builtin_amdgcn_wmma_f32_16x16x4_f32(false, a, false, b,
                                                     (short)0, acc[n],
                                                     false, false);
    }
  }
}

// =====================================================================
// Kernel 2: pass 1 — compute m per edge tile, accumulate per-channel
// sum / sum-of-squares (LDS reduce, then per-block global atomics).
// =====================================================================
__global__ void k_edge_pass1(const float* __restrict__ ef,
                             const float* __restrict__ Wedge,
                             const float* __restrict__ bedge,
                             const float* __restrict__ Hcat,
                             const int* __restrict__ src,
                             const int* __restrict__ dst,
                             float* __restrict__ estats) {
  __shared__ float lds_wt[CH * 34];
  __shared__ float red[256];

  stage_wedge(Wedge, lds_wt);
  red[threadIdx.x] = 0.f;
  __syncthreads();

  const int lane = threadIdx.x & 31;
  const int wave = threadIdx.x >> 5;
  const int wid  = blockIdx.x * 8 + wave;     // 0 .. 99999
  const int ebase = wid * 16;
  const int lo = lane & 15;
  const int hi = lane >> 4;

  v8f acc[8];
  edge_gemm_tile(ef, lds_wt, ebase, lo, hi, acc);

  float be[8];
#pragma unroll
  for (int n = 0; n < 8; ++n) be[n] = bedge[n * 16 + lo];

  float s1[8], s2[8];
#pragma unroll
  for (int n = 0; n < 8; ++n) { s1[n] = 0.f; s2[n] = 0.f; }

#pragma unroll
  for (int v = 0; v < 8; ++v) {
    const int e  = ebase + v + 8 * hi;
    const size_t so = (size_t)src[e] * 256;         // h_src row
    const size_t dofs = (size_t)dst[e] * 256 + CH;  // h_dst row (upper half)
#pragma unroll
    for (int n = 0; n < 8; ++n) {
      const int c = n * 16 + lo;
      const float m = acc[n][v] + Hcat[so + c] + Hcat[dofs + c] + be[n];
      s1[n] += m;
      s2[n] += m * m;
    }
  }
#pragma unroll
  for (int n = 0; n < 8; ++n) {
    const int c = n * 16 + lo;
    atomicAdd(&red[c], s1[n]);          // ds_add_f32
    atomicAdd(&red[CH + c], s2[n]);
  }
  __syncthreads();
  atomicAdd(&estats[threadIdx.x], red[threadIdx.x]);   // one global atomic/thread
}

// =====================================================================
// Kernel 3: finalize edge-BN: scale = gamma * rsqrt(var+eps),
//           shift = beta - mean*scale
// =====================================================================
__global__ void k_edge_finalize(const float* __restrict__ estats,
                                const float* __restrict__ gamma,
                                const float* __restrict__ beta,
                                float* __restrict__ ebn) {
  const int c = threadIdx.x;
  if (c < CH) {
    const float inv = 1.f / (float)N_EDGES;
    const float mean = estats[c] * inv;
    const float var  = fmaxf(estats[CH + c] * inv - mean * mean, 0.f);
    const float sc = gamma[c] * rsqrtf(var + BN_EPS);
    ebn[c]      = sc;
    ebn[CH + c] = beta[c] - mean * sc;
  }
}

// =====================================================================
// Kernel 4: pass 2 — recompute m (gathers hit L2), apply BN,
// gate = sigmoid(m[:, :64]) * softplus(m[:, 64:]), segment-sum into acc.
// =====================================================================
__global__ void k_edge_pass2(const float* __restrict__ ef,
                             const float* __restrict__ Wedge,
                             const float* __restrict__ bedge,
                             const float* __restrict__ Hcat,
                             const int* __restrict__ src,
                             const int* __restrict__ dst,
                             const float* __restrict__ ebn,
                             float* __restrict__ acc_node) {
  __shared__ float lds_wt[CH * 34];
  stage_wedge(Wedge, lds_wt);
  __syncthreads();

  const int lane = threadIdx.x & 31;
  const int wave = threadIdx.x >> 5;
  const int wid  = blockIdx.x * 8 + wave;
  const int ebase = wid * 16;
  const int lo = lane & 15;
  const int hi = lane >> 4;

  v8f acc[8];
  edge_gemm_tile(ef, lds_wt, ebase, lo, hi, acc);

  float be[8], sc[8], sh[8];
#pragma unroll
  for (int n = 0; n < 8; ++n) {
    const int c = n * 16 + lo;
    be[n] = bedge[c];
    sc[n] = ebn[c];
    sh[n] = ebn[CH + c];
  }

#pragma unroll
  for (int v = 0; v < 8; ++v) {
    const int e  = ebase + v + 8 * hi;
    const int d  = dst[e];
    const size_t so = (size_t)src[e] * 256;
    const size_t dofs = (size_t)d * 256 + CH;
    float mh[8];
#pragma unroll
    for (int n = 0; n < 8; ++n) {
      const int c = n * 16 + lo;
      const float m = acc[n][v] + Hcat[so + c] + Hcat[dofs + c] + be[n];
      mh[n] = m * sc[n] + sh[n];                   // batchnorm
    }
#pragma unroll
    for (int n = 0; n < 4; ++n) {                  // cols c and c+64 pair up
      const float g = sigmoid_f(mh[n]) * softplus_f(mh[n + 4]);
      atomicAdd(&acc_node[(size_t)d * NF + n * 16 + lo], g);  // segment_sum
    }
  }
}

// =====================================================================
// Kernel 5: per-channel node stats over acc (64 channels).
// blockDim = 64: thread c owns channel c, grid-strides over rows.
// =====================================================================
__global__ void k_node_stats(const float* __restrict__ acc_node,
                             float* __restrict__ nstats) {
  const int c = threadIdx.x;
  float s1 = 0.f, s2 = 0.f;
  for (int r = blockIdx.x; r < N_NODES; r += gridDim.x) {
    const float x = acc_node[(size_t)r * NF + c];
    s1 += x;
    s2 += x * x;
  }
  atomicAdd(&nstats[c], s1);
  atomicAdd(&nstats[NF + c], s2);
}

__global__ void k_node_finalize(const float* __restrict__ nstats,
                                const float* __restrict__ gamma,
                                const float* __restrict__ beta,
                                float* __restrict__ nbn) {
  const int c = threadIdx.x;
  if (c < NF) {
    const float inv = 1.f / (float)N_NODES;
    const float mean = nstats[c] * inv;
    const float var  = fmaxf(nstats[NF + c] * inv - mean * mean, 0.f);
    const float s = gamma[c] * rsqrtf(var + BN_EPS);
    nbn[c]      = s;
    nbn[NF + c] = beta[c] - mean * s;
  }
}

// =====================================================================
// Kernel 6: out = softplus(node_feats + BN(acc))
// =====================================================================
__global__ void k_output(const float* __restrict__ nf,
                         const float* __restrict__ acc_node,
                         const float* __restrict__ nbn,
                         float* __restrict__ out) {
  const size_t i = (size_t)blockIdx.x * blockDim.x + threadIdx.x;
  if (i < (size_t)N_NODES * NF) {
    const int c = (int)(i & (NF - 1));
    const float h = acc_node[i] * nbn[c] + nbn[NF + c];
    out[i] = softplus_f(nf[i] + h);
  }
}

// =====================================================================
extern "C" void kernel_launch(void* const* d_in, const int* in_sizes, int n_in,
                              void* d_out, int out_size, void* d_ws, size_t ws_size,
                              hipStream_t stream) {
  (void)in_sizes; (void)n_in; (void)out_size; (void)ws_size;
  const float* nf    = (const float*)d_in[0];
  const float* ef    = (const float*)d_in[1];
  const float* Wsrc  = (const float*)d_in[2];
  const float* bsrc  = (const float*)d_in[3];
  const float* Wdst  = (const float*)d_in[4];
  const float* bdst  = (const float*)d_in[5];
  const float* Wedge = (const float*)d_in[6];
  const float* bedge = (const float*)d_in[7];
  const float* gm    = (const float*)d_in[8];
  const float* bm    = (const float*)d_in[9];
  const float* gn    = (const float*)d_in[10];
  const float* bn    = (const float*)d_in[11];
  const int*   src   = (const int*)d_in[12];
  const int*   dst   = (const int*)d_in[13];
  float* out = (float*)d_out;

  char* ws = (char*)d_ws;
  float* Hcat   = (float*)(ws + OFF_HCAT);
  float* accN   = (float*)(ws + OFF_ACC);
  float* estats = (float*)(ws + OFF_ESTAT);
  float* ebn    = (float*)(ws + OFF_EBN);
  float* nstats = (float*)(ws + OFF_NSTAT);
  float* nbn    = (float*)(ws + OFF_NBN);

  // zero accumulators (graph-capture-safe async memsets)
  (void)hipMemsetAsync(estats, 0, SZ_ESTAT, stream);
  (void)hipMemsetAsync(accN,   0, SZ_ACC,   stream);
  (void)hipMemsetAsync(nstats, 0, SZ_NSTAT, stream);

  // 1) node GEMMs -> Hcat    (100000 waves, 8 waves/block)
  k_node_gemm<<<12500, 256, 0, stream>>>(nf, Wsrc, bsrc, Wdst, bdst, Hcat);

  // 2) edge pass 1: BN statistics   (1.6M edges / 16 per wave)
  k_edge_pass1<<<12500, 256, 0, stream>>>(ef, Wedge, bedge, Hcat, src, dst, estats);

  // 3) finalize edge BN scale/shift
  k_edge_finalize<<<1, 128, 0, stream>>>(estats, gm, bm, ebn);

  // 4) edge pass 2: normalize, gate, segment-sum into accN
  k_edge_pass2<<<12500, 256, 0, stream>>>(ef, Wedge, bedge, Hcat, src, dst, ebn, accN);

  // 5) node BN stats + finalize
  k_node_stats<<<512, 64, 0, stream>>>(accN, nstats);
  k_node_finalize<<<1, 64, 0, stream>>>(nstats, gn, bn, nbn);

  // 6) output
  k_output<<<(N_NODES * NF + 255) / 256, 256, 0, stream>>>(nf, accN, nbn, out);
}